// FusedMoEWithLoRA_7791070675062
// MI455X (gfx1250) — compile-verified
//
#include <hip/hip_runtime.h>

// Problem constants (from reference): E,H,I,K,L,R,T
constexpr int E_ = 8;
constexpr int H_ = 2048;
constexpr int I_ = 1024;
constexpr int TWOI_ = 2 * I_;
constexpr int K_ = 2;
constexpr int L_ = 4;
constexpr int R_ = 16;
constexpr int T_ = 4096;
constexpr int MROWS = T_ * K_;          // 8192 gathered (token, slot) rows
constexpr int MCAP  = MROWS + E_ * 16;  // padded capacity (each expert padded to 16)

typedef __attribute__((ext_vector_type(2))) float v2f;
typedef __attribute__((ext_vector_type(8))) float v8f;

__device__ __forceinline__ v8f wmma_f32x4(v2f a, v2f b, v8f c) {
  // V_WMMA_F32_16X16X4_F32: D(16x16,f32) = A(16x4,f32) x B(4x16,f32) + C
  return __builtin_amdgcn_wmma_f32_16x16x4_f32(false, a, false, b, (short)0, c,
                                               false, false);
}

// CDNA5 async memory->LDS copy (16B per enabled lane), tracked with ASYNCcnt.
// global_load_async_to_lds_b128 VDST(=LDS byte addr), VADDR(=64b global addr)
__device__ __forceinline__ void async_copy_b128(void* lds_dst,
                                                const void* gsrc) {
  unsigned lds =
      (unsigned)(uintptr_t)(__attribute__((address_space(3))) void*)lds_dst;
  unsigned long long ga = (unsigned long long)(uintptr_t)gsrc;
  asm volatile("global_load_async_to_lds_b128 %0, %1, off" ::"v"(lds), "v"(ga)
               : "memory");
}
__device__ __forceinline__ void wait_async0() {
  asm volatile("s_wait_asynccnt 0" ::: "memory");
}

// ---------------------------------------------------------------------------
// Kernel 0: init output, row tables, counters
// ---------------------------------------------------------------------------
__global__ void k_init(float* __restrict__ out, int* __restrict__ row_token,
                       float* __restrict__ row_w, int* __restrict__ row_lora,
                       int* __restrict__ cnt, int* __restrict__ cur) {
  const size_t stride = (size_t)gridDim.x * blockDim.x;
  for (size_t i = (size_t)blockIdx.x * blockDim.x + threadIdx.x;
       i < (size_t)T_ * H_; i += stride)
    out[i] = 0.f;
  const int i = blockIdx.x * blockDim.x + threadIdx.x;
  if (i < MCAP) {
    row_token[i] = -1;
    row_w[i] = 0.f;
    row_lora[i] = 0;
  }
  if (i < E_) {
    cnt[i] = 0;
    cur[i] = 0;
  }
}

// ---------------------------------------------------------------------------
// Kernel 1: count rows per expert
// ---------------------------------------------------------------------------
__global__ void k_count(const int* __restrict__ ids, int* __restrict__ cnt) {
  const int i = blockIdx.x * blockDim.x + threadIdx.x;
  if (i >= MROWS) return;
  atomicAdd(&cnt[ids[i]], 1);
}

// ---------------------------------------------------------------------------
// Kernel 2: exclusive offsets, padded to 16-row tiles
// ---------------------------------------------------------------------------
__global__ void k_offsets(const int* __restrict__ cnt, int* __restrict__ off) {
  if (threadIdx.x == 0 && blockIdx.x == 0) {
    int a = 0;
    for (int e = 0; e < E_; ++e) {
      off[e] = a;
      a += ((cnt[e] + 15) >> 4) << 4;
    }
    off[E_] = a;
  }
}

// ---------------------------------------------------------------------------
// Kernel 3: scatter (t,k) slots into expert-grouped row table
// ---------------------------------------------------------------------------
__global__ void k_scatter(const int* __restrict__ ids,
                          const float* __restrict__ tw,
                          const int* __restrict__ lidx,
                          const int* __restrict__ off, int* __restrict__ cur,
                          int* __restrict__ row_token, float* __restrict__ row_w,
                          int* __restrict__ row_lora) {
  const int i = blockIdx.x * blockDim.x + threadIdx.x;
  if (i >= MROWS) return;
  const int e = ids[i];
  const int pos = off[e] + atomicAdd(&cur[e], 1);
  const int t = i / K_;
  row_token[pos] = t;
  row_w[pos] = tw[i];
  row_lora[pos] = lidx[t];
}

// ---------------------------------------------------------------------------
// Kernel 4: grouped gate_up GEMM + LoRA#1 + SwiGLU -> act (gathered rows x I)
//   One workgroup (8 waves) per 16-row tile. Each wave produces paired
//   16x16 gate/up tiles via v_wmma_f32_16x16x4_f32 over K=H. The X tile is
//   staged into LDS with GLOBAL_LOAD_ASYNC_TO_LDS_B128 (ASYNCcnt).
// ---------------------------------------------------------------------------
__global__ __launch_bounds__(256) void k_gateup(
    const float* __restrict__ x, const float* __restrict__ w13,
    const float* __restrict__ gua, const float* __restrict__ gub,
    const int* __restrict__ row_token, const int* __restrict__ row_lora,
    const int* __restrict__ off, float* __restrict__ act) {
  __shared__ float Xs[16][H_ + 4];  // +4 pad: conflict-free ds_load_b64 A frags
  __shared__ float Us[16][16];      // LoRA u = x . A^T
  __shared__ int lors[16];
  __shared__ int toks[16];
  __shared__ int s_e;

  const int row0 = blockIdx.x * 16;
  if (threadIdx.x == 0) {
    int e = -1;
    if (row0 < off[E_]) {
      for (int i = 0; i < E_; ++i)
        if (row0 >= off[i] && row0 < off[i + 1]) { e = i; break; }
    }
    s_e = e;
  }
  if (threadIdx.x < 16) {
    const int tok = row_token[row0 + threadIdx.x];
    toks[threadIdx.x] = (tok < 0) ? 0 : tok;  // padded rows read row 0
    lors[threadIdx.x] = (tok >= 0) ? row_lora[row0 + threadIdx.x] : 0;
  }
  __syncthreads();
  const int e = s_e;
  if (e < 0) return;

  // Stage the 16 x H X-tile into LDS via async copies (no VGPR round-trip).
  for (int idx = threadIdx.x; idx < 16 * (H_ / 4); idx += 256) {
    const int m = idx / (H_ / 4);
    const int c4 = idx % (H_ / 4);
    async_copy_b128(&Xs[m][c4 * 4], x + (size_t)toks[m] * H_ + c4 * 4);
  }
  wait_async0();
  __syncthreads();

  // LoRA-A: u[i][r] = sum_h X[i][h] * gu_a[l_i, e, r, h]   (16x16 values)
  {
    const int i = threadIdx.x >> 4, r = threadIdx.x & 15;
    const float* A = gua + (((size_t)lors[i] * E_ + e) * R_ + r) * H_;
    float acc = 0.f;
#pragma unroll 4
    for (int h = 0; h < H_; ++h) acc += Xs[i][h] * A[h];
    Us[i][r] = acc;
  }
  __syncthreads();

  const int wave = threadIdx.x >> 5;
  const int lane = threadIdx.x & 31;
  const int half = lane >> 4;
  const int n = lane & 15;   // B/C fragment column for this lane
  const int ma = lane & 15;  // A fragment row for this lane

  for (int jt = wave * 16; jt < I_; jt += 8 * 16) {
    v8f cg = {};
    v8f cu = {};
    const float* wg = w13 + ((size_t)e * TWOI_ + (jt + n)) * H_;
    const float* wu = w13 + ((size_t)e * TWOI_ + I_ + (jt + n)) * H_;
    __builtin_prefetch(wg, 0, 0);
    __builtin_prefetch(wu, 0, 0);
#pragma unroll 4
    for (int k = 0; k < H_; k += 4) {
      const int kk = k + 2 * half;  // lane-half selects K{0,1} vs K{2,3}
      v2f a = *(const v2f*)&Xs[ma][kk];
      v2f bg; bg.x = wg[kk]; bg.y = wg[kk + 1];
      v2f bu; bu.x = wu[kk]; bu.y = wu[kk + 1];
      cg = wmma_f32x4(a, bg, cg);
      cu = wmma_f32x4(a, bu, cu);
    }
    // LoRA-B delta (per-row adapter), SwiGLU, store act
#pragma unroll
    for (int v = 0; v < 8; ++v) {
      const int m = v + 8 * half;  // C fragment row for VGPR v
      const int l = lors[m];
      const float* bgv = gub + (((size_t)l * E_ + e) * TWOI_ + (jt + n)) * R_;
      const float* buv =
          gub + (((size_t)l * E_ + e) * TWOI_ + (I_ + jt + n)) * R_;
      float dg = 0.f, du = 0.f;
#pragma unroll
      for (int r = 0; r < R_; ++r) {
        const float uu = Us[m][r];
        dg += uu * bgv[r];
        du += uu * buv[r];
      }
      const float hg = cg[v] + dg;
      const float hu = cu[v] + du;
      const float sg = hg / (1.f + __expf(-hg));  // silu
      act[(size_t)(row0 + m) * I_ + (jt + n)] = sg * hu;
    }
  }
}

// ---------------------------------------------------------------------------
// Kernel 5: grouped down GEMM + LoRA#2 + weighted combine into out
// ---------------------------------------------------------------------------
__global__ __launch_bounds__(256) void k_down(
    const float* __restrict__ act, const float* __restrict__ w2,
    const float* __restrict__ dna, const float* __restrict__ dnb,
    const int* __restrict__ row_token, const float* __restrict__ row_w,
    const int* __restrict__ row_lora, const int* __restrict__ off,
    float* __restrict__ out) {
  __shared__ float As[16][I_ + 4];
  __shared__ float Vs[16][16];
  __shared__ int lors[16];
  __shared__ int toks[16];
  __shared__ float wts[16];
  __shared__ int s_e;

  const int row0 = blockIdx.x * 16;
  if (threadIdx.x == 0) {
    int e = -1;
    if (row0 < off[E_]) {
      for (int i = 0; i < E_; ++i)
        if (row0 >= off[i] && row0 < off[i + 1]) { e = i; break; }
    }
    s_e = e;
  }
  if (threadIdx.x < 16) {
    const int tok = row_token[row0 + threadIdx.x];
    toks[threadIdx.x] = tok;
    wts[threadIdx.x] = row_w[row0 + threadIdx.x];
    lors[threadIdx.x] = (tok >= 0) ? row_lora[row0 + threadIdx.x] : 0;
  }
  __syncthreads();
  const int e = s_e;
  if (e < 0) return;

  // Stage the 16 x I act-tile into LDS via async copies.
  for (int idx = threadIdx.x; idx < 16 * (I_ / 4); idx += 256) {
    const int m = idx / (I_ / 4);
    const int c4 = idx % (I_ / 4);
    async_copy_b128(&As[m][c4 * 4], act + (size_t)(row0 + m) * I_ + c4 * 4);
  }
  wait_async0();
  __syncthreads();

  // LoRA-A (down): v[i][r] = sum_k act[i][k] * d_a[l_i, e, r, k]
  {
    const int i = threadIdx.x >> 4, r = threadIdx.x & 15;
    const float* A = dna + (((size_t)lors[i] * E_ + e) * R_ + r) * I_;
    float acc = 0.f;
#pragma unroll 4
    for (int k = 0; k < I_; ++k) acc += As[i][k] * A[k];
    Vs[i][r] = acc;
  }
  __syncthreads();

  const int wave = threadIdx.x >> 5;
  const int lane = threadIdx.x & 31;
  const int half = lane >> 4;
  const int n = lane & 15;
  const int ma = lane & 15;

  for (int jt = wave * 16; jt < H_; jt += 8 * 16) {
    v8f c = {};
    const float* wr = w2 + ((size_t)e * H_ + (jt + n)) * I_;
    __builtin_prefetch(wr, 0, 0);
#pragma unroll 4
    for (int k = 0; k < I_; k += 4) {
      const int kk = k + 2 * half;
      v2f a = *(const v2f*)&As[ma][kk];
      v2f b; b.x = wr[kk]; b.y = wr[kk + 1];
      c = wmma_f32x4(a, b, c);
    }
#pragma unroll
    for (int v = 0; v < 8; ++v) {
      const int m = v + 8 * half;
      const int tok = toks[m];
      if (tok < 0) continue;  // padded row
      const int l = lors[m];
      const float* bb = dnb + (((size_t)l * E_ + e) * H_ + (jt + n)) * R_;
      float dd = 0.f;
#pragma unroll
      for (int r = 0; r < R_; ++r) dd += Vs[m][r] * bb[r];
      const float val = wts[m] * (c[v] + dd);
      unsafeAtomicAdd(&out[(size_t)tok * H_ + (jt + n)], val);
    }
  }
}

// ---------------------------------------------------------------------------
// Host launcher
// ---------------------------------------------------------------------------
extern "C" void kernel_launch(void* const* d_in, const int* in_sizes, int n_in,
                              void* d_out, int out_size, void* d_ws,
                              size_t ws_size, hipStream_t stream) {
  const float* hidden = (const float*)d_in[0];
  const float* topk_w = (const float*)d_in[1];
  const float* w13 = (const float*)d_in[2];
  const float* w2 = (const float*)d_in[3];
  const float* gua = (const float*)d_in[4];
  const float* gub = (const float*)d_in[5];
  const float* dna = (const float*)d_in[6];
  const float* dnb = (const float*)d_in[7];
  const int* ids = (const int*)d_in[8];
  const int* lidx = (const int*)d_in[9];
  float* out = (float*)d_out;

  char* p = (char*)d_ws;
  auto alloc = [&](size_t bytes) {
    void* q = (void*)p;
    p += (bytes + 255) & ~(size_t)255;
    return q;
  };
  int* cnt = (int*)alloc(E_ * sizeof(int));
  int* off = (int*)alloc((E_ + 1) * sizeof(int));
  int* cur = (int*)alloc(E_ * sizeof(int));
  int* row_token = (int*)alloc(MCAP * sizeof(int));
  int* row_lora = (int*)alloc(MCAP * sizeof(int));
  float* row_w = (float*)alloc(MCAP * sizeof(float));
  float* act = (float*)alloc((size_t)MCAP * I_ * sizeof(float));

  k_init<<<4096, 256, 0, stream>>>(out, row_token, row_w, row_lora, cnt, cur);
  k_count<<<MROWS / 256, 256, 0, stream>>>(ids, cnt);
  k_offsets<<<1, 1, 0, stream>>>(cnt, off);
  k_scatter<<<MROWS / 256, 256, 0, stream>>>(ids, topk_w, lidx, off, cur,
                                             row_token, row_w, row_lora);
  k_gateup<<<MCAP / 16, 256, 0, stream>>>(hidden, w13, gua, gub, row_token,
                                          row_lora, off, act);
  k_down<<<MCAP / 16, 256, 0, stream>>>(act, w2, dna, dnb, row_token, row_w,
                                        row_lora, off, out);
}